// HeteroDotProductPredictor_28217935134750
// MI455X (gfx1250) — compile-verified
//
#include <hip/hip_runtime.h>

typedef __attribute__((ext_vector_type(2))) float v2f;
typedef __attribute__((ext_vector_type(8))) float v8f;

#define D_FEAT          128
#define ROW_PAD         132   // row stride in floats: 128 + 4 pad -> all 64 LDS banks hit
#define EDGES_PER_WAVE  16
#define WAVES_PER_BLOCK 2
#define THREADS         (WAVES_PER_BLOCK * 32)

// One wave computes 16 edge scores:
//   A (16x128) = h[src[e0..e15]], B (128x16) = h[dst[e0..e15]] columns
//   D = A x B accumulated via 32 x V_WMMA_F32_16X16X4_F32 ; score = diag(D)
__global__ __launch_bounds__(THREADS)
void edge_dot_wmma(const float* __restrict__ h,
                   const long long* __restrict__ src,
                   const long long* __restrict__ dst,
                   float* __restrict__ out,
                   long long n_edges)
{
    __shared__ float smem[WAVES_PER_BLOCK * 32 * ROW_PAD];

    const int lane = threadIdx.x & 31;
    const int wave = threadIdx.x >> 5;
    float* ws = smem + wave * 32 * ROW_PAD;

    const long long tile  = (long long)blockIdx.x * WAVES_PER_BLOCK + wave;
    const long long ebase = tile * EDGES_PER_WAVE;

    // Lanes 0..15 fetch src indices for edges 0..15; lanes 16..31 fetch dst indices.
    long long e = ebase + (long long)(lane & 15);
    if (e >= n_edges) e = n_edges - 1;
    const long long myrow = (lane < 16) ? src[e] : dst[e];

    // Stage 32 gathered rows into LDS; each iteration is one fully-coalesced
    // 512B row read (32 lanes x float4) from global (L2-resident: h = 51MB < 192MB).
    for (int r = 0; r < 32; ++r) {
        long long row = __shfl(myrow, r, 32);
        const float4* rp = (const float4*)(h + row * D_FEAT);
        float4 v = rp[lane];
        *(float4*)(ws + r * ROW_PAD + lane * 4) = v;
    }
    // Same-wave LDS ops are in-order (DScnt); no cross-wave sharing -> no barrier.

    // WMMA operand addressing (identical pattern for A and B):
    //   16x4 f32 A: lanes 0-15 -> {K+0,K+1} of row (lane&15); lanes 16-31 -> {K+2,K+3}
    //   4x16 f32 B: same per-lane pair, taken from the dst rows (rows 16..31)
    const int half = lane >> 4;          // 0 or 1
    const int m    = lane & 15;
    const float* aBase = ws + m * ROW_PAD + half * 2;        // src rows 0..15
    const float* bBase = aBase + 16 * ROW_PAD;               // dst rows 16..31

    v8f acc = {};
#pragma unroll
    for (int kk = 0; kk < 32; ++kk) {
        v2f a = *(const v2f*)(aBase + kk * 4);   // ds_load_b64, conflict-free
        v2f b = *(const v2f*)(bBase + kk * 4);   // ds_load_b64, conflict-free
        acc = __builtin_amdgcn_wmma_f32_16x16x4_f32(
            /*neg_a=*/false, a, /*neg_b=*/false, b,
            /*c_mod=*/(short)0, acc, /*reuse_a=*/false, /*reuse_b=*/false);
    }

    // Extract diag(D). C/D layout: VGPR r holds M=r (lanes 0-15, N=lane) and
    // M=r+8 (lanes 16-31, N=lane-16). Diagonal M==N lives at:
    //   m<8 : lane m,    vgpr m
    //   m>=8: lane m+16, vgpr m-8
    // Spill acc to the (now dead) wave-private LDS region and gather.
#pragma unroll
    for (int i = 0; i < 8; ++i)
        ws[lane * 8 + i] = acc[i];

    if (lane < 16) {
        const int sl = (lane < 8) ? lane : (lane + 16);
        const int sr = (lane < 8) ? lane : (lane - 8);
        const float score = ws[sl * 8 + sr];
        const long long eo = ebase + lane;
        if (eo < n_edges) out[eo] = score;
    }
}

extern "C" void kernel_launch(void* const* d_in, const int* in_sizes, int n_in,
                              void* d_out, int out_size, void* d_ws, size_t ws_size,
                              hipStream_t stream)
{
    const float*     h   = (const float*)d_in[0];
    const long long* src = (const long long*)d_in[1];
    const long long* dst = (const long long*)d_in[2];
    float*           out = (float*)d_out;

    const long long n_edges = (long long)in_sizes[1];
    const long long tiles   = (n_edges + EDGES_PER_WAVE - 1) / EDGES_PER_WAVE;
    const long long blocks  = (tiles + WAVES_PER_BLOCK - 1) / WAVES_PER_BLOCK;

    edge_dot_wmma<<<(unsigned)blocks, THREADS, 0, stream>>>(h, src, dst, out, n_edges);
}